// SoftLengthRegulator_46248207843459
// MI455X (gfx1250) — compile-verified
//
#include <hip/hip_runtime.h>
#include <hip/hip_bf16.h>

// ---------------- constants (match reference) ----------------
#define BB   8
#define KK_  256
#define DD   256
#define TT   2048
#define NH_  4
#define HD_  64
#define FW_  8
#define SW_  16
#define SC_  2

typedef float v2f __attribute__((ext_vector_type(2)));
typedef float v8f __attribute__((ext_vector_type(8)));

__device__ __forceinline__ float silu_f(float x) {
    return x / (1.0f + __expf(-x));
}

// ================= K1: duration + cumsum =================
__global__ __launch_bounds__(256) void dur_kernel(
    const float* __restrict__ phs, const unsigned char* __restrict__ mask,
    const float* __restrict__ Wd, const float* __restrict__ bd,
    float* __restrict__ d_out_d, float* __restrict__ s_arr, float* __restrict__ e_arr)
{
    __shared__ float sh[KK_];
    int b = blockIdx.x, k = threadIdx.x;
    const float* row = phs + (b * KK_ + k) * DD;
    float acc = bd[0];
    #pragma unroll 4
    for (int d = 0; d < DD; ++d) acc += row[d] * Wd[d];
    float dl = fmaxf(acc, 0.0f);                 // relu
    float dd = fmaxf(__expf(dl) - 1.0f, 1e-12f); // clip(exp-1, 1e-12)
    if (mask[b * KK_ + k] == 0) dd = 0.0f;       // v_pad -> 0
    d_out_d[b * KK_ + k] = dd;
    sh[k] = dd;
    __syncthreads();
    // inclusive scan (Hillis-Steele)
    for (int off = 1; off < KK_; off <<= 1) {
        float prev = (k >= off) ? sh[k - off] : 0.0f;
        __syncthreads();
        sh[k] += prev;
        __syncthreads();
    }
    float ee = sh[k];
    e_arr[b * KK_ + k] = ee;
    s_arr[b * KK_ + k] = ee - dd;
}

// ================= K2: v = phs @ W1 + b1 (f32 WMMA) =================
// M = B*K = 2048, N = 256, K = 256. 1 wave per 16x16 tile, 8 waves/block.
__global__ __launch_bounds__(256) void gemm_v_kernel(
    const float* __restrict__ A, const float* __restrict__ W,
    const float* __restrict__ bias, float* __restrict__ C)
{
    int tid  = threadIdx.x;
    int wave = blockIdx.x * 8 + (tid >> 5);
    int tileM = wave >> 4;        // 0..127
    int tileN = wave & 15;        // 0..15
    int lane = tid & 31, l15 = lane & 15, hi = lane >> 4;
    int rowBase = tileM * 16, colBase = tileN * 16;
    const float* arow = A + (rowBase + l15) * DD;
    v8f acc = {0.f,0.f,0.f,0.f,0.f,0.f,0.f,0.f};
    #pragma unroll 4
    for (int k0 = 0; k0 < DD; k0 += 4) {
        int ka = k0 + 2 * hi;
        v2f a, bf;
        a.x  = arow[ka];
        a.y  = arow[ka + 1];
        bf.x = W[ka * DD + colBase + l15];
        bf.y = W[(ka + 1) * DD + colBase + l15];
        acc = __builtin_amdgcn_wmma_f32_16x16x4_f32(false, a, false, bf,
                                                    (short)0, acc, false, false);
    }
    int col = colBase + l15;
    float bv = bias[col];
    #pragma unroll
    for (int g = 0; g < 8; ++g)
        C[(rowBase + g + 8 * hi) * DD + col] = acc[g] + bv;
}

// ================= K3: conv1d(3)+LN(F=8)+SiLU, both paths =================
// grid = B * (K/16); 128 threads: kl = tid>>3 (16 k), f = tid&7.
__global__ __launch_bounds__(128) void conv_kernel(
    const float* __restrict__ v,
    const float* __restrict__ wk, const float* __restrict__ wb,
    const float* __restrict__ wg, const float* __restrict__ wbeta,
    const float* __restrict__ ck, const float* __restrict__ cb,
    const float* __restrict__ cg, const float* __restrict__ cbeta,
    float* __restrict__ wfeat, float* __restrict__ cfeat)
{
    __shared__ float vsh[18 * DD];
    int b  = blockIdx.x >> 4;
    int kt = blockIdx.x & 15;
    int k0 = kt * 16;
    int tid = threadIdx.x;
    for (int i = tid; i < 18 * DD; i += 128) {
        int r = i >> 8, d = i & 255;
        int kk = k0 - 1 + r;
        vsh[i] = (kk >= 0 && kk < KK_) ? v[(b * KK_ + kk) * DD + d] : 0.0f;
    }
    __syncthreads();
    int kl = tid >> 3, f = tid & 7;
    float yw = wb[f], yc = cb[f];
    for (int d = 0; d < DD; ++d) {
        float v0 = vsh[(kl + 0) * DD + d];
        float v1 = vsh[(kl + 1) * DD + d];
        float v2 = vsh[(kl + 2) * DD + d];
        const float* wkp = wk + (f * DD + d) * 3;
        const float* ckp = ck + (f * DD + d) * 3;
        yw += v0 * wkp[0] + v1 * wkp[1] + v2 * wkp[2];
        yc += v0 * ckp[0] + v1 * ckp[1] + v2 * ckp[2];
    }
    // LayerNorm over the 8 channels: 8-lane butterfly (wave32-safe, groups aligned)
    float sw = yw, sww = yw * yw, sc = yc, scc = yc * yc;
    for (int off = 1; off < 8; off <<= 1) {
        sw  += __shfl_xor(sw,  off, 32);
        sww += __shfl_xor(sww, off, 32);
        sc  += __shfl_xor(sc,  off, 32);
        scc += __shfl_xor(scc, off, 32);
    }
    float muw = sw * 0.125f, varw = sww * 0.125f - muw * muw;
    float muc = sc * 0.125f, varc = scc * 0.125f - muc * muc;
    float nw = (yw - muw) * rsqrtf(varw + 1e-5f) * wg[f] + wbeta[f];
    float nc = (yc - muc) * rsqrtf(varc + 1e-5f) * cg[f] + cbeta[f];
    int k = k0 + kl;
    wfeat[(b * KK_ + k) * FW_ + f] = silu_f(nw);
    cfeat[(b * KK_ + k) * FW_ + f] = silu_f(nc);
}

// ================= K4: per-(b,t) MLPs + softmax + ein + c_extra =================
__global__ __launch_bounds__(256) void attn_kernel(
    const unsigned char* __restrict__ mask, const int* __restrict__ dur,
    const float* __restrict__ s_arr, const float* __restrict__ e_arr,
    const float* __restrict__ wfeat, const float* __restrict__ cfeat,
    const float* __restrict__ sw1W, const float* __restrict__ sw1b,
    const float* __restrict__ sw2W, const float* __restrict__ sw2b,
    const float* __restrict__ weW,  const float* __restrict__ web,
    const float* __restrict__ sc1W, const float* __restrict__ sc1b,
    const float* __restrict__ sc2W, const float* __restrict__ sc2b,
    const float* __restrict__ ceW,  const float* __restrict__ ceb,
    float* __restrict__ attn_out, float* __restrict__ c_flat)
{
    __shared__ float Llog[4 * KK_];
    __shared__ float Lhc[2 * KK_];
    __shared__ float red[512];
    __shared__ float wA[160], bA[16], wB[256], bB[16], wE[64], bE[4];
    __shared__ float wC1[20], bC1[2], wC2[4], bC2[2], wCE[128], bCE[64];
    __shared__ float mxs[4], sms[4], eins[8];

    int tid = threadIdx.x;
    if (tid < 160) wA[tid] = sw1W[tid];
    if (tid < 16) { bA[tid] = sw1b[tid]; bB[tid] = sw2b[tid]; }
    wB[tid] = sw2W[tid];
    if (tid < 64) { wE[tid] = weW[tid]; bCE[tid] = ceb[tid]; }
    if (tid < 4)  { bE[tid] = web[tid]; wC2[tid] = sc2W[tid]; }
    if (tid < 20) wC1[tid] = sc1W[tid];
    if (tid < 2)  { bC1[tid] = sc1b[tid]; bC2[tid] = sc2b[tid]; }
    if (tid < 128) wCE[tid] = ceW[tid];
    int b = blockIdx.x >> 11;   // T = 2048
    int t = blockIdx.x & (TT - 1);
    __syncthreads();

    // ---- phase A: per-k MLPs ----
    int k = tid;
    bool vp = (mask[b * KK_ + k] == 0);
    bool am = vp || (t >= dur[b]);
    float scale = (float)(t + 1);
    float m[10];
    m[0] = am ? 0.0f : scale - s_arr[b * KK_ + k];
    m[1] = am ? 0.0f : e_arr[b * KK_ + k] - scale;
    #pragma unroll
    for (int f = 0; f < 8; ++f) m[2 + f] = wfeat[(b * KK_ + k) * FW_ + f];

    float h1[SW_];
    #pragma unroll
    for (int j = 0; j < SW_; ++j) {
        float a = bA[j];
        #pragma unroll
        for (int i = 0; i < 10; ++i) a += m[i] * wA[i * SW_ + j];
        h1[j] = silu_f(a);
    }
    float h2[SW_];
    #pragma unroll
    for (int j = 0; j < SW_; ++j) {
        float a = bB[j];
        #pragma unroll
        for (int i = 0; i < SW_; ++i) a += h1[i] * wB[i * SW_ + j];
        h2[j] = silu_f(a);
    }
    #pragma unroll
    for (int h = 0; h < NH_; ++h) {
        float a = bE[h];
        #pragma unroll
        for (int j = 0; j < SW_; ++j) a += h2[j] * wE[j * NH_ + h];
        Llog[h * KK_ + k] = vp ? -1e30f : a;   // mask uses v_pad only (ref)
    }
    // c path (reuse m[0..1], replace conv features)
    #pragma unroll
    for (int f = 0; f < 8; ++f) m[2 + f] = cfeat[(b * KK_ + k) * FW_ + f];
    float c1[2];
    #pragma unroll
    for (int j = 0; j < 2; ++j) {
        float a = bC1[j];
        #pragma unroll
        for (int i = 0; i < 10; ++i) a += m[i] * wC1[i * 2 + j];
        c1[j] = silu_f(a);
    }
    #pragma unroll
    for (int j = 0; j < 2; ++j) {
        float a = bC2[j] + c1[0] * wC2[j] + c1[1] * wC2[2 + j];
        Lhc[j * KK_ + k] = silu_f(a);
    }
    __syncthreads();

    // ---- phase B: max over k per head (4 segs x 64) ----
    {
        int h = tid >> 6, j = tid & 63;
        float mx = Llog[h * KK_ + j];
        mx = fmaxf(mx, Llog[h * KK_ + j + 64]);
        mx = fmaxf(mx, Llog[h * KK_ + j + 128]);
        mx = fmaxf(mx, Llog[h * KK_ + j + 192]);
        red[h * 64 + j] = mx;
    }
    __syncthreads();
    for (int s = 32; s >= 1; s >>= 1) {
        int h = tid >> 6, j = tid & 63;
        if (j < s) red[h * 64 + j] = fmaxf(red[h * 64 + j], red[h * 64 + j + s]);
        __syncthreads();
    }
    if (tid < 4) mxs[tid] = red[tid * 64];
    __syncthreads();

    // ---- phase C: exp + sum ----
    {
        int h = tid >> 6, j = tid & 63;
        float mx = mxs[h], ssum = 0.0f;
        #pragma unroll
        for (int mm = 0; mm < 4; ++mm) {
            int kk = j + 64 * mm;
            float p = __expf(Llog[h * KK_ + kk] - mx);
            Llog[h * KK_ + kk] = p;
            ssum += p;
        }
        red[h * 64 + j] = ssum;
    }
    __syncthreads();
    for (int s = 32; s >= 1; s >>= 1) {
        int h = tid >> 6, j = tid & 63;
        if (j < s) red[h * 64 + j] += red[h * 64 + j + s];
        __syncthreads();
    }
    if (tid < 4) sms[tid] = red[tid * 64];
    __syncthreads();

    // ---- phase D: normalize + write attn output ----
    #pragma unroll
    for (int h = 0; h < NH_; ++h) {
        float a = Llog[h * KK_ + k] / sms[h];
        Llog[h * KK_ + k] = a;
        attn_out[(((b * NH_ + h) * TT + t) * KK_) + k] = a;
    }
    __syncthreads();

    // ---- phase E: ein[h][p] = sum_k attn*hc (8 segs x 64) ----
    {
        int h = tid >> 6, j = tid & 63;
        float p0 = 0.0f, p1 = 0.0f;
        #pragma unroll
        for (int mm = 0; mm < 4; ++mm) {
            int kk = j + 64 * mm;
            float a = Llog[h * KK_ + kk];
            p0 += a * Lhc[kk];
            p1 += a * Lhc[KK_ + kk];
        }
        red[(h * 2 + 0) * 64 + j] = p0;
        red[(h * 2 + 1) * 64 + j] = p1;
    }
    __syncthreads();
    for (int s = 32; s >= 1; s >>= 1) {
        int seg = tid >> 5, j = tid & 31;
        if (j < s) red[seg * 64 + j] += red[seg * 64 + j + s];
        __syncthreads();
    }
    if (tid < 8) eins[tid] = red[tid * 64];
    __syncthreads();

    // ---- phase F: c_extra = ein @ ceW + ceb, flattened (b,t, h*64+d) ----
    {
        int h = tid >> 6, dl = tid & 63;
        float val = bCE[dl] + eins[h * 2 + 0] * wCE[dl] + eins[h * 2 + 1] * wCE[64 + dl];
        c_flat[(b * TT + t) * DD + h * HD_ + dl] = val;
    }
}

// ================= K5: o_pre = attn @ v4 per (b,h) (f32 WMMA) =================
// per (b,h): M=2048(t), N=64, K=256 -> 512 tiles; 32 pairs -> 16384 waves.
__global__ __launch_bounds__(256) void gemm_oleft_kernel(
    const float* __restrict__ attn, const float* __restrict__ v,
    float* __restrict__ o_pre)
{
    int tid  = threadIdx.x;
    int wave = blockIdx.x * 8 + (tid >> 5);
    int bh   = wave >> 9;          // /512
    int tile = wave & 511;
    int tileM = tile >> 2, tileN = tile & 3;
    int b = bh >> 2, h = bh & 3;
    const float* A  = attn + ((size_t)(b * NH_ + h) * TT) * KK_;
    const float* Bm = v + b * KK_ * DD + h * HD_;
    float* C = o_pre + (size_t)b * TT * DD + h * HD_;
    int lane = tid & 31, l15 = lane & 15, hi = lane >> 4;
    int rowBase = tileM * 16, colBase = tileN * 16;
    const float* arow = A + (rowBase + l15) * KK_;
    v8f acc = {0.f,0.f,0.f,0.f,0.f,0.f,0.f,0.f};
    #pragma unroll 4
    for (int k0 = 0; k0 < KK_; k0 += 4) {
        int ka = k0 + 2 * hi;
        v2f a, bf;
        a.x  = arow[ka];
        a.y  = arow[ka + 1];
        bf.x = Bm[ka * DD + colBase + l15];
        bf.y = Bm[(ka + 1) * DD + colBase + l15];
        acc = __builtin_amdgcn_wmma_f32_16x16x4_f32(false, a, false, bf,
                                                    (short)0, acc, false, false);
    }
    #pragma unroll
    for (int g = 0; g < 8; ++g)
        C[(rowBase + g + 8 * hi) * DD + colBase + l15] = acc[g];
}

// ================= K6: out = o_pre@W2 + c_flat@W3 + b2 + b3 (f32 WMMA) =================
// M = B*T = 16384, N = 256, effective K = 512 (two concatenated GEMMs).
__global__ __launch_bounds__(256) void gemm_out_kernel(
    const float* __restrict__ A1, const float* __restrict__ W2,
    const float* __restrict__ A2, const float* __restrict__ W3,
    const float* __restrict__ b2, const float* __restrict__ b3,
    float* __restrict__ out)
{
    int tid  = threadIdx.x;
    int wave = blockIdx.x * 8 + (tid >> 5);
    int tileM = wave >> 4;        // 0..1023
    int tileN = wave & 15;
    int lane = tid & 31, l15 = lane & 15, hi = lane >> 4;
    int rowBase = tileM * 16, colBase = tileN * 16;
    v8f acc = {0.f,0.f,0.f,0.f,0.f,0.f,0.f,0.f};
    const float* arow1 = A1 + (size_t)(rowBase + l15) * DD;
    const float* arow2 = A2 + (size_t)(rowBase + l15) * DD;
    #pragma unroll 4
    for (int k0 = 0; k0 < DD; k0 += 4) {
        int ka = k0 + 2 * hi;
        v2f a, bf;
        a.x  = arow1[ka];
        a.y  = arow1[ka + 1];
        bf.x = W2[ka * DD + colBase + l15];
        bf.y = W2[(ka + 1) * DD + colBase + l15];
        acc = __builtin_amdgcn_wmma_f32_16x16x4_f32(false, a, false, bf,
                                                    (short)0, acc, false, false);
    }
    #pragma unroll 4
    for (int k0 = 0; k0 < DD; k0 += 4) {
        int ka = k0 + 2 * hi;
        v2f a, bf;
        a.x  = arow2[ka];
        a.y  = arow2[ka + 1];
        bf.x = W3[ka * DD + colBase + l15];
        bf.y = W3[(ka + 1) * DD + colBase + l15];
        acc = __builtin_amdgcn_wmma_f32_16x16x4_f32(false, a, false, bf,
                                                    (short)0, acc, false, false);
    }
    int col = colBase + l15;
    float bv = b2[col] + b3[col];
    #pragma unroll
    for (int g = 0; g < 8; ++g)
        out[(size_t)(rowBase + g + 8 * hi) * DD + col] = acc[g] + bv;
}

// ================= launch =================
extern "C" void kernel_launch(void* const* d_in, const int* in_sizes, int n_in,
                              void* d_out, int out_size, void* d_ws, size_t ws_size,
                              hipStream_t stream) {
    (void)in_sizes; (void)n_in; (void)out_size; (void)ws_size;
    // inputs (setup_inputs dict order)
    const float*         phs    = (const float*)d_in[0];
    const unsigned char* pmask  = (const unsigned char*)d_in[1];
    const int*           dur    = (const int*)d_in[2];
    // d_in[3] = T (constant 2048, hardcoded)
    const float* Wd  = (const float*)d_in[4];
    const float* bd  = (const float*)d_in[5];
    const float* W1  = (const float*)d_in[6];
    const float* b1  = (const float*)d_in[7];
    const float* W2  = (const float*)d_in[8];
    const float* b2  = (const float*)d_in[9];
    const float* W3  = (const float*)d_in[10];
    const float* b3  = (const float*)d_in[11];
    const float* wconv_k = (const float*)d_in[12];
    const float* wconv_b = (const float*)d_in[13];
    const float* wln_g   = (const float*)d_in[14];
    const float* wln_b   = (const float*)d_in[15];
    const float* cconv_k = (const float*)d_in[16];
    const float* cconv_b = (const float*)d_in[17];
    const float* cln_g   = (const float*)d_in[18];
    const float* cln_b   = (const float*)d_in[19];
    const float* sw1_W = (const float*)d_in[20];
    const float* sw1_b = (const float*)d_in[21];
    const float* sw2_W = (const float*)d_in[22];
    const float* sw2_b = (const float*)d_in[23];
    const float* sc1_W = (const float*)d_in[24];
    const float* sc1_b = (const float*)d_in[25];
    const float* sc2_W = (const float*)d_in[26];
    const float* sc2_b = (const float*)d_in[27];
    const float* we_W  = (const float*)d_in[28];
    const float* we_b  = (const float*)d_in[29];
    const float* ce_W  = (const float*)d_in[30];
    const float* ce_b  = (const float*)d_in[31];

    // outputs: (out, d, attn) concatenated
    float* out_p  = (float*)d_out;                       // B*T*D = 4194304
    float* d_p    = out_p + (size_t)BB * TT * DD;        // B*K   = 2048
    float* attn_p = d_p + (size_t)BB * KK_;              // B*NH*T*K = 16777216

    // workspace layout (floats)
    float* ws = (float*)d_ws;
    float* v_p     = ws;                                    // B*K*D   = 524288
    float* s_p     = v_p + (size_t)BB * KK_ * DD;           // B*K
    float* e_p     = s_p + (size_t)BB * KK_;                // B*K
    float* wfeat_p = e_p + (size_t)BB * KK_;                // B*K*8
    float* cfeat_p = wfeat_p + (size_t)BB * KK_ * FW_;      // B*K*8
    float* opre_p  = cfeat_p + (size_t)BB * KK_ * FW_;      // B*T*256 = 4194304
    float* cflat_p = opre_p + (size_t)BB * TT * DD;         // B*T*256 = 4194304

    // K1: duration + cumsum
    dur_kernel<<<BB, 256, 0, stream>>>(phs, pmask, Wd, bd, d_p, s_p, e_p);
    // K2: v = phs @ W1 + b1   (2048x256x256, WMMA f32)
    gemm_v_kernel<<<256, 256, 0, stream>>>(phs, W1, b1, v_p);
    // K3: conv+LN+SiLU features
    conv_kernel<<<BB * 16, 128, 0, stream>>>(v_p,
        wconv_k, wconv_b, wln_g, wln_b,
        cconv_k, cconv_b, cln_g, cln_b,
        wfeat_p, cfeat_p);
    // K4: softmax attention + ein + c_extra
    attn_kernel<<<BB * TT, 256, 0, stream>>>(pmask, dur, s_p, e_p,
        wfeat_p, cfeat_p,
        sw1_W, sw1_b, sw2_W, sw2_b, we_W, we_b,
        sc1_W, sc1_b, sc2_W, sc2_b, ce_W, ce_b,
        attn_p, cflat_p);
    // K5: o_pre = attn @ v4   (32 GEMMs 2048x64x256, WMMA f32)
    gemm_oleft_kernel<<<2048, 256, 0, stream>>>(attn_p, v_p, opre_p);
    // K6: out = o_pre@W2 + c_flat@W3 + b2 + b3   (16384x256x512, WMMA f32)
    gemm_out_kernel<<<2048, 256, 0, stream>>>(opre_p, W2, cflat_p, W3, b2, b3, out_p);
}